// PNO4D_72559177498645
// MI455X (gfx1250) — compile-verified
//
#include <hip/hip_runtime.h>
#include <hip/hip_bf16.h>
#include <math.h>

// ---------------- problem constants ----------------
#define XD 32
#define YD 32
#define ZD 32
#define AD 16
#define NPTS (XD*YD*ZD*AD)        // 524288
#define WCH 20
#define HIDN 128
#define NL 4
#define MM1 8
#define MM2 8
#define MM3 8
#define MM4 5
#define NMODES (MM1*MM2*MM3*MM4)  // 2560

typedef __attribute__((ext_vector_type(2))) float v2f;
typedef __attribute__((ext_vector_type(8))) float v8f;

// ---------------- helpers ----------------
static __device__ __forceinline__ int bitrev_c(int x, int bits) {
  int r = 0;
  for (int i = 0; i < bits; ++i) { r = (r << 1) | (x & 1); x >>= 1; }
  return r;
}

// Compile-time twiddle tables: stage m=2^s, entries j=0..m/2-1 at offset (m/2 - 1).
// TWC = cos(2*pi*j/m), TWS = sin(2*pi*j/m). Forward uses (TWC, -TWS); inverse (TWC, +TWS).
__device__ constexpr float TWC[31] = {
  1.f,
  1.f, 0.f,
  1.f, 0.70710678f, 0.f, -0.70710678f,
  1.f, 0.92387953f, 0.70710678f, 0.38268343f, 0.f,
  -0.38268343f, -0.70710678f, -0.92387953f,
  1.f, 0.98078528f, 0.92387953f, 0.83146961f, 0.70710678f, 0.55557023f,
  0.38268343f, 0.19509032f, 0.f, -0.19509032f, -0.38268343f, -0.55557023f,
  -0.70710678f, -0.83146961f, -0.92387953f, -0.98078528f
};
__device__ constexpr float TWS[31] = {
  0.f,
  0.f, 1.f,
  0.f, 0.70710678f, 1.f, 0.70710678f,
  0.f, 0.38268343f, 0.70710678f, 0.92387953f, 1.f,
  0.92387953f, 0.70710678f, 0.38268343f,
  0.f, 0.19509032f, 0.38268343f, 0.55557023f, 0.70710678f, 0.83146961f,
  0.92387953f, 0.98078528f, 1.f, 0.98078528f, 0.92387953f, 0.83146961f,
  0.70710678f, 0.55557023f, 0.38268343f, 0.19509032f
};

// ---------------- FFT: one axis pass, line fully in registers ----------------
// Separable C2C FFT, direction as template param so twiddles are immediates.
template<int L, int LOG2L, int TPB, bool INV>
__launch_bounds__(TPB)
__global__ void fft_axis_kernel(float2* __restrict__ data, int nchan, int stride) {
  const int linesPerChan = NPTS / L;
  long gl = (long)blockIdx.x * TPB + threadIdx.x;
  long total = (long)nchan * linesPerChan;
  if (gl >= total) return;
  int chan = (int)(gl / linesPerChan);
  int l    = (int)(gl % linesPerChan);
  long base = (long)chan * NPTS + (long)(l / stride) * (long)(stride * L) + (l % stride);
  __builtin_prefetch(data + base, 0, 1);  // -> global_prefetch_b8

  float vr[L], vi[L];
#pragma unroll
  for (int i = 0; i < L; ++i) {
    float2 t = data[base + (long)i * stride];
    vr[i] = t.x; vi[i] = t.y;
  }
#pragma unroll
  for (int i = 0; i < L; ++i) {
    const int j = bitrev_c(i, LOG2L);
    if (j > i) {
      float tr = vr[i]; vr[i] = vr[j]; vr[j] = tr;
      float ti = vi[i]; vi[i] = vi[j]; vi[j] = ti;
    }
  }
#pragma unroll
  for (int s = 1; s <= LOG2L; ++s) {
    const int m = 1 << s, mh = m >> 1;
    const int off = mh - 1;
#pragma unroll
    for (int k = 0; k < L; k += m) {
#pragma unroll
      for (int j = 0; j < mh; ++j) {
        const float ca = TWC[off + j];
        const float sa = INV ? TWS[off + j] : -TWS[off + j];
        float tr = ca * vr[k + j + mh] - sa * vi[k + j + mh];
        float ti = ca * vi[k + j + mh] + sa * vr[k + j + mh];
        float ur = vr[k + j], ui = vi[k + j];
        vr[k + j]      = ur + tr;  vi[k + j]      = ui + ti;
        vr[k + j + mh] = ur - tr;  vi[k + j + mh] = ui - ti;
      }
    }
  }
  const float scale = INV ? (1.f / (float)L) : 1.f;
#pragma unroll
  for (int i = 0; i < L; ++i)
    data[base + (long)i * stride] = make_float2(vr[i] * scale, vi[i] * scale);
}

// ---------------- pack / init kernels ----------------
__global__ void pack_chi_kernel(const float* __restrict__ chi, float2* __restrict__ dst) {
  long p = (long)blockIdx.x * blockDim.x + threadIdx.x;
  if (p < NPTS) dst[p] = make_float2(chi[p], 0.f);
}

__global__ void pack_cx_kernel(const float* __restrict__ chi, const float* __restrict__ xbuf,
                               float2* __restrict__ dst) {
  long p = (long)blockIdx.x * blockDim.x + threadIdx.x;
  if (p >= NPTS) return;
  float c = chi[p];
#pragma unroll
  for (int ch = 0; ch < WCH; ++ch)
    dst[(long)ch * NPTS + p] = make_float2(c * xbuf[(long)ch * NPTS + p], 0.f);
}

__global__ void init_x_kernel(const float* __restrict__ fw, const float* __restrict__ fb,
                              float* __restrict__ xbuf) {
  long p = (long)blockIdx.x * blockDim.x + threadIdx.x;
  if (p >= NPTS) return;
  int ix = (int)(p >> 14);
  int iy = (int)((p >> 9) & 31);
  int iz = (int)((p >> 4) & 31);
  int ia = (int)(p & 15);
  float gx = ix * (1.f / 31.f);
  float gy = iy * (1.f / 31.f);
  float gz = iz * (1.f / 31.f);
  float ga = ia * (1.f / 15.f);
#pragma unroll
  for (int c = 0; c < WCH; ++c) {
    xbuf[(long)c * NPTS + p] =
        fb[c] + fw[c * 4 + 0] * gx + fw[c * 4 + 1] * gy +
                fw[c * 4 + 2] * gz + fw[c * 4 + 3] * ga;
  }
}

__global__ void zero_f4_kernel(float4* __restrict__ p, long n4) {
  long i = (long)blockIdx.x * blockDim.x + threadIdx.x;
  if (i < n4) p[i] = make_float4(0.f, 0.f, 0.f, 0.f);
}

// ---------------- spectral octant multiply (+ Hermitian mirror for irfftn equiv) ----------------
// Thread mapping: mpos fastest -> coalesced weight stream (65 MB/layer).
__global__ void spectral_mul_kernel(const float2* __restrict__ cxhat,
                                    const float2* __restrict__ chihat,
                                    const float2* __restrict__ scw,  // layer base
                                    float2* __restrict__ outx,
                                    float2* __restrict__ outc) {
  int tid = blockIdx.x * blockDim.x + threadIdx.x;
  const int total = 8 * NMODES * WCH;  // 409600
  if (tid >= total) return;
  int mpos = tid % NMODES;
  int rest = tid / NMODES;
  int o    = rest % WCH;
  int k    = rest / WCH;
  int i4 = mpos % MM4;
  int i3 = (mpos / MM4) % MM3;
  int i2 = (mpos / (MM4 * MM3)) % MM2;
  int i1 = mpos / (MM4 * MM3 * MM2);
  // octant hi-flags: x:{1,4,5,7}=0xB2  y:{2,4,6,7}=0xD4  z:{3,5,6,7}=0xE8
  int b1 = (0xB2 >> k) & 1;
  int b2 = (0xD4 >> k) & 1;
  int b3 = (0xE8 >> k) & 1;
  int gx = b1 ? (XD - MM1 + i1) : i1;
  int gy = b2 ? (YD - MM2 + i2) : i2;
  int gz = b3 ? (ZD - MM3 + i3) : i3;
  int gt = i4;
  long spos = ((long)(gx * YD + gy) * ZD + gz) * AD + gt;

  float2 chv = chihat[spos];
  float axr = 0.f, axi = 0.f, wsr = 0.f, wsi = 0.f;
#pragma unroll 4
  for (int i = 0; i < WCH; ++i) {
    float2 w  = scw[(((long)k * WCH + i) * WCH + o) * NMODES + mpos];
    float2 xv = cxhat[(long)i * NPTS + spos];
    axr += w.x * xv.x - w.y * xv.y;
    axi += w.x * xv.y + w.y * xv.x;
    wsr += w.x;  wsi += w.y;
  }
  float ccr = wsr * chv.x - wsi * chv.y;
  float cci = wsr * chv.y + wsi * chv.x;
  long dst = (long)o * NPTS + spos;
  outx[dst] = make_float2(axr, axi);
  outc[dst] = make_float2(ccr, cci);
  if (gt >= 1) {  // Hermitian mirror along A with XYZ negation: matches irfftn semantics
    int mx = (XD - gx) % XD, my = (YD - gy) % YD, mz = (ZD - gz) % ZD, mt = AD - gt;
    long mdst = (long)o * NPTS + ((long)(mx * YD + my) * ZD + mz) * AD + mt;
    outx[mdst] = make_float2(axr, -axi);
    outc[mdst] = make_float2(ccr, -cci);
  }
}

// ---------------- fused pointwise (20x20) + combine + GELU ----------------
__global__ void combine_kernel(const float* __restrict__ chi, float* __restrict__ xbuf,
                               const float2* __restrict__ outx, const float2* __restrict__ outc,
                               const float* __restrict__ pw, const float* __restrict__ pwb,
                               int apply_gelu) {
  long p = (long)blockIdx.x * blockDim.x + threadIdx.x;
  if (p >= NPTS) return;
  float xin[WCH];
#pragma unroll
  for (int c = 0; c < WCH; ++c) xin[c] = xbuf[(long)c * NPTS + p];
  float chv = chi[p];
#pragma unroll 2
  for (int o = 0; o < WCH; ++o) {
    float wx = pwb[o];
#pragma unroll
    for (int i = 0; i < WCH; ++i) wx += pw[o * WCH + i] * xin[i];
    float cxr = outx[(long)o * NPTS + p].x;
    float ccr = outc[(long)o * NPTS + p].x;
    float xn = chv * (cxr - xin[o] * ccr + wx);
    if (apply_gelu) xn = 0.5f * xn * (1.f + erff(xn * 0.70710678118654752f));
    xbuf[(long)o * NPTS + p] = xn;
  }
}

// ---------------- metric-head prep: Meff = wcon@u0 + u1, softplus(act_w), gather g ----------------
__global__ void prep_meff_kernel(const float* __restrict__ wc, const float* __restrict__ u0,
                                 const float* __restrict__ u1, float* __restrict__ meff) {
  int tid = blockIdx.x * blockDim.x + threadIdx.x;
  if (tid >= HIDN * WCH) return;
  int j = tid / WCH, i = tid % WCH;
  float acc = u1[j * WCH + i];
  for (int kk = 0; kk < HIDN; ++kk) {
    float w = wc[j * HIDN + kk];
    float wcon = fminf(w * w, fabsf(w));
    acc += wcon * u0[kk * WCH + i];
  }
  meff[j * WCH + i] = acc;
}

__global__ void prep_misc_kernel(const float* __restrict__ actw, const int* __restrict__ gs,
                                 const float* __restrict__ xbuf,
                                 float* __restrict__ sbuf, float* __restrict__ gvec) {
  int t = threadIdx.x;
  if (t < HIDN) sbuf[t] = log1pf(__expf(actw[t]));
  if (t < WCH) {
    int ix = gs[0], iy = gs[1], iz = gs[2], ia = gs[3];
    long gp = ((long)(ix * YD + iy) * ZD + iz) * AD + ia;
    gvec[t] = xbuf[(long)t * NPTS + gp];
  }
}

// ---------------- metric head: v = (x - g) @ Meff^T  via V_WMMA_F32_16X16X4_F32 ----------------
// One wave owns a 16-point tile; 8 N-tiles x 5 K-steps = 40 wmma per wave.
// A layout (ISA 7.12.2): lanes 0-15 hold M rows at K=k0,k0+1; lanes 16-31 at K=k0+2,k0+3.
// koff = lane>=16 ? 2 : 0 is folded into load ADDRESSES (no per-element cndmask ladders,
// and each lane loads only the 10 K-values it actually feeds into WMMA).
// D layout: lane<16 -> N=lane, M=r; lane>=16 -> N=lane-16, M=r+8.
__launch_bounds__(128)
__global__ void metric_wmma_kernel(const float* __restrict__ xbuf,
                                   const float* __restrict__ meff,
                                   const float* __restrict__ gvec,
                                   const float* __restrict__ sbuf,
                                   float* __restrict__ out) {
  __shared__ float vs[4][16][HIDN];   // 32 KB
  __shared__ float smeff[HIDN * WCH]; // 10 KB
  __shared__ float ssb[HIDN];         // 0.5 KB
  // stage Meff + softplus(act_w) once per block
  for (int i = threadIdx.x; i < HIDN * WCH; i += 128) smeff[i] = meff[i];
  if (threadIdx.x < HIDN) ssb[threadIdx.x] = sbuf[threadIdx.x];
  __syncthreads();

  const int wave = threadIdx.x >> 5;
  const int lane = threadIdx.x & 31;
  const int row  = lane & 15;
  const bool lo  = (lane < 16);
  const int koff = lo ? 0 : 2;
  const long p = (long)blockIdx.x * 64 + wave * 16 + row;

  // A fragments: lane-half-specific K values, loaded directly at k0+koff
  float a0[5], a1[5];
#pragma unroll
  for (int s = 0; s < 5; ++s) {
    const int kA = 4 * s + koff;
    a0[s] = xbuf[(long)kA * NPTS + p] - gvec[kA];
    a1[s] = xbuf[(long)(kA + 1) * NPTS + p] - gvec[kA + 1];
  }

#pragma unroll
  for (int nt = 0; nt < 8; ++nt) {
    v8f acc = {0.f, 0.f, 0.f, 0.f, 0.f, 0.f, 0.f, 0.f};
    const float* mrow = smeff + (nt * 16 + row) * WCH + koff;
#pragma unroll
    for (int s = 0; s < 5; ++s) {
      v2f a, b;
      a.x = a0[s];           a.y = a1[s];
      b.x = mrow[4 * s];     b.y = mrow[4 * s + 1];
      acc = __builtin_amdgcn_wmma_f32_16x16x4_f32(false, a, false, b,
                                                  (short)0, acc, false, false);
    }
#pragma unroll
    for (int r = 0; r < 8; ++r) {
      const int m = lo ? r : (r + 8);
      vs[wave][m][nt * 16 + row] = acc[r];
    }
  }
  __syncthreads();

  // closed-form maxrelu(v)+maxrelu(-v) then final maxrelu + mean, one thread per point
  if (threadIdx.x < 64) {
    const int w2 = threadIdx.x >> 4;
    const int m2 = threadIdx.x & 15;
    float acc1 = 0.f, acc2 = 0.f;
#pragma unroll
    for (int jj = 0; jj < 32; ++jj) {
      float e0 = vs[w2][m2][4 * jj + 0];
      float e1 = vs[w2][m2][4 * jj + 1];
      float e2 = vs[w2][m2][4 * jj + 2];
      float e3 = vs[w2][m2][4 * jj + 3];
      // maxrelu(v)+maxrelu(-v): max-part -> |a-b|, relu-part -> (|a|s_a+|b|s_b)/2
      float A0 = fabsf(e0 - e1);
      float A1 = fabsf(e2 - e3);
      float B0 = 0.5f * (fabsf(e0) * ssb[4 * jj + 0] + fabsf(e1) * ssb[4 * jj + 1]);
      float B1 = 0.5f * (fabsf(e2) * ssb[4 * jj + 2] + fabsf(e3) * ssb[4 * jj + 3]);
      // final maxrelu over h2=[A(64),B(64)] (all >=0 so relu(h2*s)=h2*s)
      acc1 += fmaxf(A0, A1);
      acc2 += 0.5f * (A0 * ssb[2 * jj] + A1 * ssb[2 * jj + 1]);
      acc1 += fmaxf(B0, B1);
      acc2 += 0.5f * (B0 * ssb[64 + 2 * jj] + B1 * ssb[64 + 2 * jj + 1]);
    }
    out[(long)blockIdx.x * 64 + threadIdx.x] = (acc1 + acc2) * (1.f / 128.f);
  }
}

// ---------------- host driver ----------------
template<bool INV>
static void fft4_run(float2* buf, int nchan, hipStream_t stream) {
  {
    const int lines = nchan * (NPTS / 16);
    const int tpb = 128;
    fft_axis_kernel<16, 4, 128, INV><<<(lines + tpb - 1) / tpb, tpb, 0, stream>>>(
        buf, nchan, 1);
  }
  const int strides32[3] = {16, 512, 16384};  // Z, Y, X
  for (int a = 0; a < 3; ++a) {
    const int lines = nchan * (NPTS / 32);
    const int tpb = 64;
    fft_axis_kernel<32, 5, 64, INV><<<(lines + tpb - 1) / tpb, tpb, 0, stream>>>(
        buf, nchan, strides32[a]);
  }
}

extern "C" void kernel_launch(void* const* d_in, const int* in_sizes, int n_in,
                              void* d_out, int out_size, void* d_ws, size_t ws_size,
                              hipStream_t stream) {
  (void)in_sizes; (void)n_in; (void)out_size; (void)ws_size;
  const float*  chi  = (const float*)d_in[0];
  const int*    gs   = (const int*)d_in[1];
  const float*  fc0w = (const float*)d_in[2];
  const float*  fc0b = (const float*)d_in[3];
  const float2* scw  = (const float2*)d_in[4];   // complex64 interleaved
  const float*  pww  = (const float*)d_in[5];
  const float*  pwb  = (const float*)d_in[6];
  const float*  u0   = (const float*)d_in[7];
  const float*  u1   = (const float*)d_in[8];
  const float*  wc   = (const float*)d_in[9];
  const float*  actw = (const float*)d_in[10];
  float* out = (float*)d_out;

  // workspace layout
  float*  ws     = (float*)d_ws;
  float2* chihat = (float2*)ws;                       // NPTS complex
  float2* cxhat  = chihat + NPTS;                     // 20*NPTS complex
  float2* outx   = cxhat + (long)WCH * NPTS;          // 20*NPTS complex
  float2* outc   = outx + (long)WCH * NPTS;           // 20*NPTS complex
  float*  xbuf   = (float*)(outc + (long)WCH * NPTS); // 20*NPTS real
  float*  meff   = xbuf + (long)WCH * NPTS;           // 128*20
  float*  sbuf   = meff + HIDN * WCH;                 // 128
  float*  gvec   = sbuf + HIDN;                       // 20

  const int TPB = 256;
  const int NB  = NPTS / TPB;  // 2048

  // chi spectrum: computed once (chi_c is a broadcast of one channel)
  pack_chi_kernel<<<NB, TPB, 0, stream>>>(chi, chihat);
  fft4_run<false>(chihat, 1, stream);

  // x0 = grid @ fc0_w^T + fc0_b
  init_x_kernel<<<NB, TPB, 0, stream>>>(fc0w, fc0b, xbuf);

  for (int l = 0; l < NL; ++l) {
    pack_cx_kernel<<<NB, TPB, 0, stream>>>(chi, xbuf, cxhat);
    fft4_run<false>(cxhat, WCH, stream);

    // zero spectral outputs (outx and outc are contiguous)
    {
      long n4 = (long)(2 * 2 * WCH) * NPTS / 4;  // 80*NPTS floats -> float4
      long zb = (n4 + TPB - 1) / TPB;
      zero_f4_kernel<<<(unsigned)zb, TPB, 0, stream>>>((float4*)outx, n4);
    }
    {
      const int total = 8 * NMODES * WCH;
      spectral_mul_kernel<<<(total + TPB - 1) / TPB, TPB, 0, stream>>>(
          cxhat, chihat, scw + (long)l * 8 * WCH * WCH * NMODES, outx, outc);
    }
    fft4_run<true>(outx, WCH, stream);
    fft4_run<true>(outc, WCH, stream);

    combine_kernel<<<NB, TPB, 0, stream>>>(chi, xbuf, outx, outc,
                                           pww + l * WCH * WCH, pwb + l * WCH,
                                           (l < NL - 1) ? 1 : 0);
  }

  // metric head prep + WMMA GEMM head
  prep_meff_kernel<<<(HIDN * WCH + TPB - 1) / TPB, TPB, 0, stream>>>(wc, u0, u1, meff);
  prep_misc_kernel<<<1, 128, 0, stream>>>(actw, gs, xbuf, sbuf, gvec);
  metric_wmma_kernel<<<NPTS / 64, 128, 0, stream>>>(xbuf, meff, gvec, sbuf, out);
}